// RecurrentGCN_26018911879765
// MI455X (gfx1250) — compile-verified
//
#include <hip/hip_runtime.h>
#include <math.h>

#define F 64
#define MT_STRIDE 66   // padded row stride for transposed fused matrices (bank-conflict-free b64 LDS loads)

typedef __attribute__((ext_vector_type(2))) float v2f;
typedef __attribute__((ext_vector_type(8))) float v8f;

// ---------------------------------------------------------------------------
// Fuse gate GEMMs: MzT[j*66+k] = (Wz @ Lz_top)[k][j], cz[j] = bz@Lz_top + lz_b
// ---------------------------------------------------------------------------
__global__ void k_prep_mats(const float* __restrict__ Wz, const float* __restrict__ Lz,
                            const float* __restrict__ bz, const float* __restrict__ lz_b,
                            const float* __restrict__ Wh, const float* __restrict__ Lh,
                            const float* __restrict__ bh, const float* __restrict__ lh_b,
                            float* __restrict__ MzT, float* __restrict__ MhT,
                            float* __restrict__ cz, float* __restrict__ ch) {
  const int tid = threadIdx.x;
  for (int idx = tid; idx < 2 * F * F; idx += blockDim.x) {
    const int g = idx >> 12;          // 0 = z gate, 1 = h gate
    const int rem = idx & 4095;
    const int j = rem >> 6, k = rem & 63;
    const float* W = g ? Wh : Wz;
    const float* L = g ? Lh : Lz;     // Lz is [128,64]; rows 0..63 = top half
    float s = 0.0f;
#pragma unroll 8
    for (int i = 0; i < F; ++i) s += W[k * F + i] * L[i * F + j];
    (g ? MhT : MzT)[j * MT_STRIDE + k] = s;
  }
  for (int idx = tid; idx < 2 * F; idx += blockDim.x) {
    const int g = idx >> 6, j = idx & 63;
    const float* b = g ? bh : bz;
    const float* L = g ? Lh : Lz;
    float s = (g ? lh_b : lz_b)[j];
    for (int i = 0; i < F; ++i) s += b[i] * L[i * F + j];
    (g ? ch : cz)[j] = s;
  }
}

// ---------------------------------------------------------------------------
// Degree: deg[n] = 1 (self loop) + sum_{dst==n} w
// ---------------------------------------------------------------------------
__global__ void k_deg_init(float* __restrict__ deg, int n) {
  int i = blockIdx.x * blockDim.x + threadIdx.x;
  if (i < n) deg[i] = 1.0f;
}

__global__ void k_deg_scatter(const int* __restrict__ dst, const float* __restrict__ ew,
                              float* __restrict__ deg, int e) {
  int i = blockIdx.x * blockDim.x + threadIdx.x;
  if (i < e) unsafeAtomicAdd(&deg[dst[i]], ew[i]);
}

__global__ void k_dinv(const float* __restrict__ deg, float* __restrict__ dinv, int n) {
  int i = blockIdx.x * blockDim.x + threadIdx.x;
  if (i < n) {
    float d = deg[i];
    dinv[i] = d > 0.0f ? rsqrtf(d) : 0.0f;
  }
}

// ---------------------------------------------------------------------------
// Ax init with self-loop term: Ax[n,:] = dinv[n]^2 * x[n,:]   (float4 path)
// ---------------------------------------------------------------------------
__global__ void k_selfloop(const float* __restrict__ x, const float* __restrict__ dinv,
                           float* __restrict__ Ax, int n) {
  int i = blockIdx.x * blockDim.x + threadIdx.x;   // over n * 16 float4s
  if (i >= n * (F / 4)) return;
  int node = i >> 4;
  float dv = dinv[node];
  float c = dv * dv;
  float4 v = ((const float4*)x)[i];
  ((float4*)Ax)[i] = make_float4(v.x * c, v.y * c, v.z * c, v.w * c);
}

// ---------------------------------------------------------------------------
// One wave per edge: coalesced float2 gather of x[src], 64 f32 atomics into Ax[dst]
// ---------------------------------------------------------------------------
__global__ void k_edge_scatter(const int* __restrict__ src, const int* __restrict__ dst,
                               const float* __restrict__ ew, const float* __restrict__ x,
                               const float* __restrict__ dinv, float* __restrict__ Ax, int e) {
  int wave = (blockIdx.x * blockDim.x + threadIdx.x) >> 5;
  int lane = threadIdx.x & 31;
  if (wave >= e) return;
  int s = src[wave];
  int d = dst[wave];
  float coef = dinv[s] * ew[wave] * dinv[d];
  float2 v = ((const float2*)(x + (size_t)s * F))[lane];
  float* arow = Ax + (size_t)d * F + lane * 2;
  unsafeAtomicAdd(arow,     coef * v.x);
  unsafeAtomicAdd(arow + 1, coef * v.y);
}

// ---------------------------------------------------------------------------
// Dense fused kernel: one wave computes 16 nodes.
//   Tz = AxTile @ Mz ; Th = AxTile @ Mh    (f32 WMMA 16x16x4, 128 per tile)
//   Z = sigmoid(Tz+cz); Ht = tanh(Th+ch); out = relu((1-Z)*Ht) @ Wo + bo
// ---------------------------------------------------------------------------
__global__ void __launch_bounds__(256) k_dense(
    const float* __restrict__ Ax,
    const float* __restrict__ MzTg, const float* __restrict__ MhTg,
    const float* __restrict__ czg, const float* __restrict__ chg,
    const float* __restrict__ Wo, const float* __restrict__ bo,
    float* __restrict__ out, int n) {
  __shared__ float sMzT[F * MT_STRIDE];
  __shared__ float sMhT[F * MT_STRIDE];
  __shared__ float sCz[F], sCh[F], sWo[F];

  const int tid = threadIdx.x;
  for (int i = tid; i < F * MT_STRIDE; i += blockDim.x) {
    sMzT[i] = MzTg[i];
    sMhT[i] = MhTg[i];
  }
  for (int i = tid; i < F; i += blockDim.x) {
    sCz[i] = czg[i];
    sCh[i] = chg[i];
    sWo[i] = Wo[i];
  }
  __syncthreads();

  const int wave = tid >> 5;
  const int lane = tid & 31;
  const int tile = blockIdx.x * (blockDim.x >> 5) + wave;
  const int nb = tile * 16;
  if (nb >= n) return;                      // wave-uniform: EXEC stays all-1 for WMMA

  const int lq = lane & 15;                 // M (for A) / N (for B,D) index within 16
  const int half = lane >> 4;               // selects K pair {0,1} vs {2,3}
  int row = nb + lq;
  if (row >= n) row = n - 1;                // tail safety (rows >= n are never written)

  const v8f vzero = {0.f, 0.f, 0.f, 0.f, 0.f, 0.f, 0.f, 0.f};
  v8f accZ[4] = {vzero, vzero, vzero, vzero};
  v8f accH[4] = {vzero, vzero, vzero, vzero};

  const float* arow = Ax + (size_t)row * F;
  const int ks = half * 2;

#pragma unroll 4
  for (int kk = 0; kk < 16; ++kk) {
    const int ka = kk * 4 + ks;             // even -> 8B aligned
    v2f a = *(const v2f*)(arow + ka);       // A: lane=M, half selects K pair
#pragma unroll
    for (int nt = 0; nt < 4; ++nt) {
      const int col = nt * 16 + lq;         // B: lane=N, half selects K pair
      v2f bz2 = *(const v2f*)(&sMzT[col * MT_STRIDE + ka]);
      accZ[nt] = __builtin_amdgcn_wmma_f32_16x16x4_f32(
          false, a, false, bz2, (short)0, accZ[nt], false, false);
      v2f bh2 = *(const v2f*)(&sMhT[col * MT_STRIDE + ka]);
      accH[nt] = __builtin_amdgcn_wmma_f32_16x16x4_f32(
          false, a, false, bh2, (short)0, accH[nt], false, false);
    }
  }

  // D layout: acc[nt][i] = (M = i + half*8, N = nt*16 + lq). Gates + readout dot.
  float partial[8] = {0.f, 0.f, 0.f, 0.f, 0.f, 0.f, 0.f, 0.f};
#pragma unroll
  for (int nt = 0; nt < 4; ++nt) {
    const int col = nt * 16 + lq;
    const float zc = sCz[col], hc = sCh[col], wo = sWo[col];
#pragma unroll
    for (int i = 0; i < 8; ++i) {
      float zg = 1.0f / (1.0f + expf(-(accZ[nt][i] + zc)));
      float ht = tanhf(accH[nt][i] + hc);
      float hn = (1.0f - zg) * ht;
      partial[i] += fmaxf(hn, 0.0f) * wo;
    }
  }

  // Reduce the N dimension across each 16-lane half (masks < 16 stay in-half).
#pragma unroll
  for (int m = 1; m < 16; m <<= 1) {
#pragma unroll
    for (int i = 0; i < 8; ++i) partial[i] += __shfl_xor(partial[i], m, 32);
  }

  if (lq == 0) {                            // lane 0 -> rows 0..7, lane 16 -> rows 8..15
    const float bob = bo[0];
    const int base = nb + half * 8;
#pragma unroll
    for (int i = 0; i < 8; ++i) {
      if (base + i < n) out[base + i] = partial[i] + bob;
    }
  }
}

// ---------------------------------------------------------------------------
// Inputs (setup_inputs order): x, edge_index, edge_weight, Wz, bz, Wr, br,
// Wh, bh, Lz, lz_b, Lr, lr_b, Lh, lh_b, Wo, bo.
// R gate (5,6,11,12) is provably dead since H==0 -> skipped entirely.
// ---------------------------------------------------------------------------
extern "C" void kernel_launch(void* const* d_in, const int* in_sizes, int n_in,
                              void* d_out, int out_size, void* d_ws, size_t ws_size,
                              hipStream_t stream) {
  const float* x    = (const float*)d_in[0];
  const int*   ei   = (const int*)d_in[1];
  const float* ew   = (const float*)d_in[2];
  const float* Wz   = (const float*)d_in[3];
  const float* bz   = (const float*)d_in[4];
  const float* Wh   = (const float*)d_in[7];
  const float* bh   = (const float*)d_in[8];
  const float* Lz   = (const float*)d_in[9];
  const float* lz_b = (const float*)d_in[10];
  const float* Lh   = (const float*)d_in[13];
  const float* lh_b = (const float*)d_in[14];
  const float* Wo   = (const float*)d_in[15];
  const float* bo   = (const float*)d_in[16];
  float* out = (float*)d_out;

  const int n = in_sizes[0] / F;
  const int e = in_sizes[1] / 2;
  const int* src = ei;
  const int* dst = ei + e;

  float* ws   = (float*)d_ws;
  float* Ax   = ws;                              // n*F
  float* deg  = Ax + (size_t)n * F;              // n
  float* dinv = deg + n;                         // n
  float* MzT  = dinv + n;                        // F*MT_STRIDE
  float* MhT  = MzT + F * MT_STRIDE;             // F*MT_STRIDE
  float* cz   = MhT + F * MT_STRIDE;             // F
  float* ch   = cz + F;                          // F

  k_prep_mats<<<1, 256, 0, stream>>>(Wz, Lz, bz, lz_b, Wh, Lh, bh, lh_b, MzT, MhT, cz, ch);
  k_deg_init<<<(n + 255) / 256, 256, 0, stream>>>(deg, n);
  k_deg_scatter<<<(e + 255) / 256, 256, 0, stream>>>(dst, ew, deg, e);
  k_dinv<<<(n + 255) / 256, 256, 0, stream>>>(deg, dinv, n);
  k_selfloop<<<(n * (F / 4) + 255) / 256, 256, 0, stream>>>(x, dinv, Ax, n);
  k_edge_scatter<<<(e + 7) / 8, 256, 0, stream>>>(src, dst, ew, x, dinv, Ax, e);

  const int tiles = (n + 15) / 16;
  k_dense<<<(tiles + 7) / 8, 256, 0, stream>>>(Ax, MzT, MhT, cz, ch, Wo, bo, out, n);
}